// ParametricGate_16862041604145
// MI455X (gfx1250) — compile-verified
//
#include <hip/hip_runtime.h>
#include <math.h>

typedef __attribute__((ext_vector_type(2))) float v2f;
typedef __attribute__((ext_vector_type(8))) float v8f;

// RY(theta) single-qubit gate on [B, 2^n] f32 state, pair stride = 2^bit.
// One V_WMMA_F32_16X16X4_F32 per wave-iteration transforms 32 consecutive pairs:
//   A (16x4)  = gate block-diag [[c,-s],[s,c]] (x2), rest zero  (built once)
//   B (4x16)  = per-lane (x0, x1) of pair (base+lane)  -> fully coalesced loads
//   D rows 0..3 = (y0,y1) of pairs 0..15 / 16..31      -> repacked via ds_bpermute
__global__ __launch_bounds__(256) void ry_gate_wmma_kernel(
    const float* __restrict__ state,
    const float* __restrict__ theta,
    const int*   __restrict__ qubit_p,
    float*       __restrict__ out,
    unsigned int nbits,       // log2(dim), e.g. 20
    unsigned int numGroups)   // total pair-groups of 32
{
    const unsigned int lane = threadIdx.x & 31u;

    // Wave-uniform values pinned to SGPRs -> scalar loop, EXEC stays all-ones.
    const unsigned int waveId = (unsigned int)__builtin_amdgcn_readfirstlane(
        (int)((blockIdx.x * blockDim.x + threadIdx.x) >> 5));
    const unsigned int nWaves = (gridDim.x * blockDim.x) >> 5;

    const float t = 0.5f * theta[0];
    const float c = cosf(t);
    const float s = sinf(t);

    // pairing bit position in the flat index (reference: bit = n-1-qubit)
    const unsigned int bit      = (nbits - 1u) - (unsigned int)qubit_p[0];
    const unsigned int stride   = 1u << bit;                 // 65536 for qubit=3
    const unsigned int pairMask = (1u << (nbits - 1u)) - 1u;

    // A-matrix (16x4 f32): lanes 0-15 hold (A[M][0],A[M][1]), M=lane;
    // lanes 16-31 hold (A[M][2],A[M][3]), M=lane-16.
    // Nonzero rows: 0:[c,-s,0,0] 1:[s,c,0,0] 2:[0,0,c,-s] 3:[0,0,s,c]
    // => per half-wave, row r0=2*hi carries (c,-s) and r1=2*hi+1 carries (s,c).
    const unsigned int hi = lane >> 4;      // 0 or 1
    const unsigned int M  = lane & 15u;
    const unsigned int r0 = 2u * hi;
    const unsigned int r1 = r0 + 1u;
    float ax = 0.0f, ay = 0.0f;
    ax = (M == r0) ?  c : ax;   // branchless: chained v_cndmask
    ax = (M == r1) ?  s : ax;
    ay = (M == r0) ? -s : ay;
    ay = (M == r1) ?  c : ay;
    v2f a; a.x = ax; a.y = ay;

    const v8f cz = {};                              // C = 0 (inline)
    const int permIdx = (int)((lane & 15u) << 2);   // byte address for ds_bpermute

    for (unsigned int g = waveId; g < numGroups; g += nWaves) {
        // All of this decomposition is SGPR math (g is scalar).
        const unsigned int P    = g << 5;             // global pair base
        const unsigned int row  = P >> (nbits - 1u);  // batch row
        const unsigned int p    = P & pairMask;       // pair index within row
        const unsigned int base = (row << nbits)
                                + ((p >> bit) << (bit + 1u))
                                + (p & (stride - 1u));
        const unsigned int i0 = base + lane;          // one v_add per iteration
        const unsigned int i1 = i0 + stride;

        // B-matrix (4x16 f32): lane L = column L (lanes 16-31 = K-block 2..3),
        // which reduces to: every lane loads its own pair — contiguous 128B loads.
        v2f b;
        b.x = state[i0];
        b.y = state[i1];

        v8f d = __builtin_amdgcn_wmma_f32_16x16x4_f32(
            /*neg_a=*/false, a, /*neg_b=*/false, b,
            /*c_mod=*/(short)0, cz, /*reuse_a=*/false, /*reuse_b=*/false);

        // d[0]/d[1] lanes 0-15 = y0/y1 of pairs P..P+15
        // d[2]/d[3] lanes 0-15 = y0/y1 of pairs P+16..P+31 -> shift to lanes 16-31
        float y0hi = __int_as_float(
            __builtin_amdgcn_ds_bpermute(permIdx, __float_as_int(d[2])));
        float y1hi = __int_as_float(
            __builtin_amdgcn_ds_bpermute(permIdx, __float_as_int(d[3])));
        float y0 = (lane < 16u) ? d[0] : y0hi;
        float y1 = (lane < 16u) ? d[1] : y1hi;

        // NT stores: keep the 128 MiB input resident in the 192 MB L2 across replays
        __builtin_nontemporal_store(y0, out + i0);
        __builtin_nontemporal_store(y1, out + i1);
    }
}

extern "C" void kernel_launch(void* const* d_in, const int* in_sizes, int n_in,
                              void* d_out, int out_size, void* d_ws, size_t ws_size,
                              hipStream_t stream) {
    (void)n_in; (void)out_size; (void)d_ws; (void)ws_size;
    const float* state = (const float*)d_in[0];
    const float* theta = (const float*)d_in[1];
    const int*   qubit = (const int*)d_in[2];
    float*       out   = (float*)d_out;

    const unsigned int total = (unsigned int)in_sizes[0];   // B * dim = 32 * 2^20
    const unsigned int BATCH = 32u;                         // per reference
    const unsigned int dim   = total / BATCH;
    unsigned int nbits = 0u;
    while ((1u << nbits) < dim) ++nbits;                    // 20

    const unsigned int numPairs  = total >> 1;              // 2^24
    const unsigned int numGroups = numPairs >> 5;           // 524288 groups of 32

    const int threads = 256;                                // 8 wave32 per block
    const int blocks  = 4096;                               // 32768 waves, 16 groups each

    ry_gate_wmma_kernel<<<blocks, threads, 0, stream>>>(
        state, theta, qubit, out, nbits, numGroups);
}